// BaseLikelihoodModel_9706626090056
// MI455X (gfx1250) — compile-verified
//
#include <hip/hip_runtime.h>
#include <math.h>

// ---------------------------------------------------------------------------
// Felsenstein pruning on a perfect binary tree, S=16 states, N=65536 nodes.
//   Stage 1: batched expm of 16x16 rate matrices -> log transition matrices
//            (one wave32 per matrix, V_WMMA_F32_16X16X4_F32 chains; A-matrix
//            fragments register-resident, T staged through LDS)
//   Stage 2: level-synchronous log-sum-exp pruning (one wave per node/level)
// ---------------------------------------------------------------------------

#define NS   16          // number of states
#define LDP  17          // padded LDS row stride (gcd(17,64)=1 -> conflict free)

typedef __attribute__((ext_vector_type(2))) float v2f;
typedef __attribute__((ext_vector_type(8))) float v8f;

__device__ __forceinline__ void lds_fence() {
  // LDS ops are in-order within a wave on CDNA5; this only has to stop the
  // compiler from reordering DS ops across the fence (plus a hard wait).
  asm volatile("s_wait_dscnt 0" ::: "memory");
}

// C = A*B with A fragments already in registers, B in LDS (row-major,
// stride LDP). Four chained V_WMMA_F32_16X16X4_F32.
// A fragment kb: lane l -> row m=l&15, K = {4kb+2*half, 4kb+2*half+1}
// B fragment kb: rows K {4kb+2*half, 4kb+2*half+1}, column n = l&15
__device__ __forceinline__ v8f wave_mm16_regA(const v2f a[4], const float* B,
                                              int mn, int half) {
  v8f c = {};
#pragma unroll
  for (int kb = 0; kb < 4; ++kb) {
    const int k0 = 4 * kb + 2 * half;
    v2f b;
    b.x = B[k0 * LDP + mn];
    b.y = B[(k0 + 1) * LDP + mn];
    c = __builtin_amdgcn_wmma_f32_16x16x4_f32(false, a[kb], false, b,
                                              (short)0, c, false, false);
  }
  return c;
}

// C = A*B with both operands in LDS (used for squaring, where A == B == T).
__device__ __forceinline__ v8f wave_mm16(const float* A, const float* B,
                                         int mn, int half) {
  v2f a[4];
#pragma unroll
  for (int kb = 0; kb < 4; ++kb) {
    const int k0 = 4 * kb + 2 * half;
    a[kb].x = A[mn * LDP + k0];
    a[kb].y = A[mn * LDP + k0 + 1];
  }
  return wave_mm16_regA(a, B, mn, half);
}

// One wave per node: T = expm((Q - diag(growth)) * t) via scaling & squaring
// with an order-12 Taylor-Horner series, then log(clip(T, EPS)) to global.
__global__ __launch_bounds__(256) void expm_log_kernel(
    const float* __restrict__ branch_lens,
    const float* __restrict__ Q,
    const float* __restrict__ growth,
    float* __restrict__ log_trans,
    int N) {
  __shared__ float Rs[NS * NS];
  __shared__ float bufT[8][NS * LDP];

  const int tid = threadIdx.x;
  if (tid < NS * NS) {                 // rate matrix, once per block
    const int i = tid >> 4;
    float v = Q[tid];
    if (i == (tid & 15)) v -= growth[i];
    Rs[tid] = v;
  }
  __syncthreads();

  const int wave = tid >> 5;
  const int lane = tid & 31;
  const int half = lane >> 4;          // 0: rows 0-7 / K even pair, 1: rows 8-15
  const int mn   = lane & 15;          // row (A frag) == column (B/C/D frag)
  const int node = blockIdx.x * 8 + wave;
  if (node >= N) return;               // wave-uniform

  float* T = bufT[wave];
  float4* out4 = (float4*)(log_trans + (size_t)node * (NS * NS));
  const float EPSV    = 1e-30f;
  const float LOG_EPS = -69.07755279f; // log(1e-30)

  // each lane owns 8 contiguous output floats: elems [lane*8, lane*8+8)
  const int orow = lane >> 1;                 // element row
  const int ocol = (lane & 1) * 8;            // first element column

  const float t = branch_lens[node];
  if (t <= EPSV) {                     // identity branch (root edge)
    float w[8];
#pragma unroll
    for (int q = 0; q < 8; ++q)
      w[q] = ((ocol + q) == orow) ? 0.0f : LOG_EPS;
    out4[lane * 2 + 0] = make_float4(w[0], w[1], w[2], w[3]);
    out4[lane * 2 + 1] = make_float4(w[4], w[5], w[6], w[7]);
    return;                            // wave-uniform
  }

  // inf-norm of M = R*t (each lane sums row mn; lanes 16-31 duplicate)
  float rowsum = 0.0f;
#pragma unroll
  for (int j = 0; j < NS; ++j) rowsum += fabsf(Rs[mn * NS + j] * t);
#pragma unroll
  for (int off = 8; off >= 1; off >>= 1)
    rowsum = fmaxf(rowsum, __shfl_xor(rowsum, off, 32));

  int sexp = 0;
  const float theta = 0.25f;
  if (rowsum > theta) {
    sexp = (int)ceilf(log2f(rowsum / theta));
    if (sexp < 0) sexp = 0;
    if (sexp > 30) sexp = 30;
  }
  const float scl = exp2f((float)-sexp) * t;

  // A-matrix fragments (A = R * t / 2^s) held in registers for the whole
  // Horner recurrence: a[kb] = row mn, K = {4kb+2h, 4kb+2h+1}
  v2f a[4];
#pragma unroll
  for (int kb = 0; kb < 4; ++kb) {
    const int k0 = 4 * kb + 2 * half;
    a[kb].x = Rs[mn * NS + k0] * scl;
    a[kb].y = Rs[mn * NS + k0 + 1] * scl;
  }

  // T = I
#pragma unroll
  for (int v = 0; v < 8; ++v) {
    const int r = v + 8 * half;
    T[r * LDP + mn] = (r == mn) ? 1.0f : 0.0f;
  }
  lds_fence();

  // Taylor-Horner order 12: T <- I + (A*T)/k.  Fully unrolled so 1/k is a
  // compile-time literal; A fragments never touch LDS.
#pragma unroll
  for (int k = 12; k >= 1; --k) {
    v8f c = wave_mm16_regA(a, T, mn, half);
    const float inv = 1.0f / (float)k;
#pragma unroll
    for (int v = 0; v < 8; ++v) {
      const int r = v + 8 * half;
      T[r * LDP + mn] = c[v] * inv + ((r == mn) ? 1.0f : 0.0f);
    }
    lds_fence();
  }

  // repeated squaring: T <- T*T
  for (int i = 0; i < sexp; ++i) {
    v8f c = wave_mm16(T, T, mn, half);
#pragma unroll
    for (int v = 0; v < 8; ++v) T[(v + 8 * half) * LDP + mn] = c[v];
    lds_fence();
  }

  // log(clip(T, EPS)) -> global; repack so each lane stores 8 contiguous
  // floats => two fully-coalesced global_store_b128 per lane.
  float r8[8];
#pragma unroll
  for (int q = 0; q < 8; ++q)
    r8[q] = __logf(fmaxf(T[orow * LDP + ocol + q], EPSV));
  out4[lane * 2 + 0] = make_float4(r8[0], r8[1], r8[2], r8[3]);
  out4[lane * 2 + 1] = make_float4(r8[4], r8[5], r8[6], r8[7]);
}

// One wave per node at a tree level. Lanes 0-15: left-child contribution,
// lanes 16-31: right-child contribution (+ log growth). Combine via shuffle.
__global__ __launch_bounds__(256) void prune_level_kernel(
    const int* __restrict__ children,
    const float* __restrict__ log_trans,
    const float* __restrict__ growth,
    float* __restrict__ partials,
    int base, int count) {
  const int wave = threadIdx.x >> 5;
  const int lane = threadIdx.x & 31;
  const int half = lane >> 4;
  const int p    = lane & 15;
  const int idx  = blockIdx.x * 8 + wave;
  if (idx >= count) return;            // wave-uniform
  const int node  = base + idx;
  const int child = children[node * 2 + half];

  float res = 0.0f;                    // right==-1 -> contribution 0 (root)
  if (child >= 0) {
    const float4* lt4 = (const float4*)(log_trans + (size_t)child * (NS * NS) + p * NS);
    const float4* lp4 = (const float4*)(partials + (size_t)child * NS);
    float v[NS];
#pragma unroll
    for (int q = 0; q < 4; ++q) {      // 4 + 4 global_load_b128, 64B aligned
      const float4 a = lt4[q];
      const float4 b = lp4[q];
      v[4 * q + 0] = a.x + b.x;
      v[4 * q + 1] = a.y + b.y;
      v[4 * q + 2] = a.z + b.z;
      v[4 * q + 3] = a.w + b.w;
    }
    float mx = v[0];
#pragma unroll
    for (int c = 1; c < NS; ++c) mx = fmaxf(mx, v[c]);
    float sum = 0.0f;
#pragma unroll
    for (int c = 0; c < NS; ++c) sum += __expf(v[c] - mx);
    res = mx + __logf(sum);
    if (half == 1) res += __logf(growth[p]);
  }
  const float other = __shfl_xor(res, 16, 32);
  if (half == 0) partials[(size_t)node * NS + p] = res + other;
}

__global__ void copy_root_kernel(const float* __restrict__ partials,
                                 float* __restrict__ out, int root) {
  const int j = threadIdx.x;
  if (j < NS) out[j] = partials[(size_t)root * NS + j];
}

extern "C" void kernel_launch(void* const* d_in, const int* in_sizes, int n_in,
                              void* d_out, int out_size, void* d_ws, size_t ws_size,
                              hipStream_t stream) {
  // setup_inputs order: postorder, children, parents, branch_lens,
  //                     init_partials, Q, levels, growth_rates
  const int*   children      = (const int*)d_in[1];
  const float* branch_lens   = (const float*)d_in[3];
  const float* init_partials = (const float*)d_in[4];
  const float* Q             = (const float*)d_in[5];
  const float* growth        = (const float*)d_in[7];

  const int N = in_sizes[0];           // total nodes (= 2L)
  const int L = N / 2;                 // leaves
  int depth = 0;
  while ((1 << depth) < L) ++depth;    // log2(L)

  float* log_trans = (float*)d_ws;                         // N*256 floats
  float* partials  = (float*)d_ws + (size_t)N * (NS * NS); // N*16 floats

  // seed partials with the leaf one-hots (internal entries get overwritten
  // at their level before ever being read)
  hipMemcpyAsync(partials, init_partials, (size_t)N * NS * sizeof(float),
                 hipMemcpyDeviceToDevice, stream);

  // Stage 1: all log transition matrices
  expm_log_kernel<<<(N + 7) / 8, 256, 0, stream>>>(branch_lens, Q, growth,
                                                   log_trans, N);

  // Stage 2: level-synchronous pruning. Perfect tree: level h occupies
  // [offs[h], offs[h]+L>>h), offs[1]=L, offs[h]=offs[h-1]+(L>>(h-1)).
  int off = L;
  for (int h = 1; h <= depth; ++h) {
    const int cnt = L >> h;
    prune_level_kernel<<<(cnt + 7) / 8, 256, 0, stream>>>(
        children, log_trans, growth, partials, off, cnt);
    off += cnt;
  }
  // unifurcating root (node N-1, right child = -1)
  prune_level_kernel<<<1, 256, 0, stream>>>(children, log_trans, growth,
                                            partials, N - 1, 1);

  copy_root_kernel<<<1, 32, 0, stream>>>(partials, (float*)d_out, N - 1);
}